// GeometryAwareAssign_17076789969267
// MI455X (gfx1250) — compile-verified
//
#include <hip/hip_runtime.h>
#include <hip/hip_bf16.h>
#include <math.h>

// Problem constants (match reference; img_w/img_h scalars are fixed by the reference file).
#define B_   8
#define N_   2000
#define M_   24
#define S_   72
#define D_   78
#define IMGW 800.0f
#define IMGH 320.0f
#define INVALID_ -100000.0f
#define BIGC 100000000.0f

typedef float v2f __attribute__((ext_vector_type(2)));
typedef float v8f __attribute__((ext_vector_type(8)));

// ---------------------------------------------------------------------------
// K1: per-(b,m) target sampled xs + meta (gx, gy, theta/90, valid)
// ---------------------------------------------------------------------------
__global__ void k_gt(const float* __restrict__ tgt, const int* __restrict__ masks,
                     float* __restrict__ gt_xs, float* __restrict__ gmeta) {
  int bm = blockIdx.x;            // b*M + m
  int s  = threadIdx.x;           // 0..71
  const float* t = tgt + (size_t)bm * D_;
  float t2 = t[2], t3 = t[3], t4 = t[4];
  float tn = tanf(t4 * 0.017453292519943295f);
  tn = fminf(fmaxf(tn, -1000.f), 1000.f);
  if (s < S_) {
    float ys = (float)s * (1.0f / 71.0f) * (IMGH - 1.0f);
    float dx = t[6 + s];
    float x = t3 + (t2 - ys) * tn + dx;
    if (dx < -10000.f) x = INVALID_;
    gt_xs[(size_t)bm * S_ + s] = x;
  }
  if (s == 0) {
    float* g = gmeta + (size_t)bm * 4;
    g[0] = t3 / (IMGW - 1.f);
    g[1] = t2 / (IMGH - 1.f);
    g[2] = t4 / 90.f;
    g[3] = masks[bm] ? 1.f : 0.f;
  }
}

// ---------------------------------------------------------------------------
// K2: per-(b,n) pred sampled xs (range-masked) + meta (cls_cost, px, py, theta/90)
// ---------------------------------------------------------------------------
__global__ void k_pred(const float* __restrict__ preds,
                       float* __restrict__ cur_pred, float* __restrict__ pmeta) {
  int bn = blockIdx.x;            // b*N + n
  int s  = threadIdx.x;
  const float* p = preds + (size_t)bn * D_;
  float p2 = p[2], p3 = p[3], p4 = p[4], p5 = p[5];
  float tn = tanf(p4 * 0.017453292519943295f);
  tn = fminf(fmaxf(tn, -1000.f), 1000.f);
  float start_idx = (1.f - p2 / (IMGH - 1.f)) * (float)(S_ - 1);
  float len_idx   = p5 / IMGH * (float)(S_ - 1);
  if (s < S_) {
    float ys = (float)s * (1.0f / 71.0f) * (IMGH - 1.f);
    float x  = p3 + (p2 - ys) * tn + p[6 + s];
    float fi = (float)s;
    bool in  = (fi >= start_idx) && (fi <= start_idx + len_idx);
    cur_pred[(size_t)bn * S_ + s] = in ? x : INVALID_;
  }
  if (s == 0) {
    float l0 = p[0], l1 = p[1];
    float score = 1.f / (1.f + expf(l0 - l1));   // softmax(...)[1]
    score = fmaxf(score, 1e-8f);
    float* q = pmeta + (size_t)bn * 4;
    q[0] = -logf(score);
    q[1] = p3 / (IMGW - 1.f);
    q[2] = p2 / (IMGH - 1.f);
    q[3] = p4 / 90.f;
  }
}

// ---------------------------------------------------------------------------
// K3a: pairwise distance via V_WMMA_F32_16X16X4_F32.
//   D = A(16x4) x B(4x16) + C, with A=(px,py), B=(-2gx,-2gy), C=|p|^2+|g|^2
//   => D[i][j] = dist^2(pred i, target j). One wave per 16-pred tile, 2 m-tiles.
//   A layout (ISA 7.12.2): lanes 0-15 rows M, v0=K0 v1=K1; lanes 16-31 K2/K3 (zero).
//   C/D layout: vgpr r, lanes 0-15 -> row r, lanes 16-31 -> row r+8; col = lane&15.
// ---------------------------------------------------------------------------
__global__ void k_dist_wmma(const float* __restrict__ pmeta, const float* __restrict__ gmeta,
                            float* __restrict__ partial_nm) {
  int blk   = blockIdx.x;
  int b     = blk / (N_ / 16);
  int ntile = blk % (N_ / 16);
  int nbase = ntile * 16;
  int lane  = threadIdx.x;

  __shared__ float spn2[16], scls[16], spth[16];
  __shared__ float sgx[32], sgy[32], sgm2[32], sgth[32], svld[32];

  v2f a;
  if (lane < 16) {
    const float* q = pmeta + ((size_t)b * N_ + nbase + lane) * 4;
    float cls = q[0], px = q[1], py = q[2], pth = q[3];
    scls[lane] = cls; spth[lane] = pth;
    spn2[lane] = px * px + py * py;
    a.x = px; a.y = py;          // K=0,1 for row `lane`
  } else {
    a.x = 0.f; a.y = 0.f;        // K=2,3 padding rows of A
  }
  {
    int m = lane;                // 0..31, zero-padded past M_=24
    float gx = 0.f, gy = 0.f, gth = 0.f, vld = 0.f;
    if (m < M_) {
      const float* g = gmeta + ((size_t)b * M_ + m) * 4;
      gx = g[0]; gy = g[1]; gth = g[2]; vld = g[3];
    }
    sgx[m] = gx; sgy[m] = gy; sgm2[m] = gx * gx + gy * gy;
    sgth[m] = gth; svld[m] = vld;
  }
  __syncthreads();

  int colh   = lane & 15;
  int rowoff = (lane >= 16) ? 8 : 0;

  for (int mt = 0; mt < 2; ++mt) {          // uniform loop; EXEC full at WMMA
    int mbase = mt * 16;
    int col   = mbase + colh;
    v2f bb;
    if (lane < 16) { bb.x = -2.f * sgx[col]; bb.y = -2.f * sgy[col]; }  // B rows K=0,1
    else           { bb.x = 0.f;             bb.y = 0.f;             }  // B rows K=2,3
    float gm2c = sgm2[col], gthc = sgth[col], vldc = svld[col];

    v8f c;
#pragma unroll
    for (int r = 0; r < 8; ++r) c[r] = spn2[r + rowoff] + gm2c;

    // 8 args: (neg_a, A, neg_b, B, c_mod, C, reuse_a, reuse_b)
    v8f d = __builtin_amdgcn_wmma_f32_16x16x4_f32(
        false, a, false, bb, (short)0, c, false, false);

    if (col < M_) {
#pragma unroll
      for (int r = 0; r < 8; ++r) {
        int rl = r + rowoff;
        float d2   = fmaxf(d[r], 0.f) + 1e-8f;
        float dist = sqrtf(d2);
        float th   = fabsf(spth[rl] - gthc);
        // 4*cls + 5*dist + 10*theta + 2 + 1e5*!valid  (IoU term added in K3b)
        float part = 4.f * scls[rl] + 5.f * dist + 10.f * th + 2.f +
                     (vldc > 0.5f ? 0.f : 100000.f);
        partial_nm[((size_t)b * N_ + nbase + rl) * M_ + col] = part;
      }
    }
  }
}

// ---------------------------------------------------------------------------
// K3b: line-IoU over S=72 + final cost = partial - 2*iou. m varies fastest so
// the 72-float pred row is broadcast across 24 consecutive lanes; gt rows fit L0.
// ---------------------------------------------------------------------------
__global__ void k_iou_cost(const float* __restrict__ cur_pred, const float* __restrict__ gt_xs,
                           const float* __restrict__ gmeta, const float* __restrict__ partial_nm,
                           float* __restrict__ cost_nm, float* __restrict__ iou_nm) {
  int flat = blockIdx.x * blockDim.x + threadIdx.x;   // (b*N + n)*M + m
  if (flat >= B_ * N_ * M_) return;
  int m  = flat % M_;
  int bn = flat / M_;
  int b  = bn / N_;
  const float* pr = cur_pred + (size_t)bn * S_;
  const float* gr = gt_xs + ((size_t)b * M_ + m) * S_;
  float osum = 0.f, usum = 0.f;
  for (int s = 0; s < S_; ++s) {
    float p = pr[s], g = gr[s];
    bool inv = (p < INVALID_ + 1.f) || (g < INVALID_ + 1.f);
    float mn = fminf(p, g), mx = fmaxf(p, g);
    if (!inv) { osum += (mn - mx + 30.f); usum += (mx - mn + 30.f); }
  }
  float iou = osum / (usum + 1e-9f);
  iou *= gmeta[((size_t)b * M_ + m) * 4 + 3];
  iou_nm[flat]  = iou;
  cost_nm[flat] = partial_nm[flat] - 2.f * iou;
}

// ---------------------------------------------------------------------------
// K4: per (b,m): dyn_k = clip(int(top-10 IoU sum),1,10) (IoU<=1 => dyn_k<=10),
// then dyn_k-pass stable min-extraction of cost -> member list. Tie order
// (value, then index) exactly matches argsort(argsort(...)) ranks.
// ---------------------------------------------------------------------------
__global__ void k_select(const float* __restrict__ iou_nm, const float* __restrict__ cost_nm,
                         const float* __restrict__ gmeta,
                         int* __restrict__ mem_idx, float* __restrict__ mem_cost,
                         int* __restrict__ mem_cnt) {
  const int T = 128;
  int bm  = blockIdx.x;           // b*M + m
  int b   = bm / M_, m = bm % M_;
  int tid = threadIdx.x;
  __shared__ float sv[T]; __shared__ int si[T];
  __shared__ float s_pv; __shared__ int s_pi; __shared__ float s_sum; __shared__ int s_k;
  if (tid == 0) { s_pv = 3.0e38f; s_pi = -1; s_sum = 0.f; }
  __syncthreads();

  // Phase 1: 10 max-extraction passes over IoU (desc value, asc index).
  for (int t = 0; t < 10; ++t) {
    float pv = s_pv; int pi = s_pi;
    float bv = 0.f;  int bi = -1;
    for (int i = tid; i < N_; i += T) {
      float v = iou_nm[((size_t)b * N_ + i) * M_ + m];
      bool elig = (v < pv) || (v == pv && i > pi);          // strictly after prev key
      if (elig && (bi < 0 || v > bv || (v == bv && i < bi))) { bv = v; bi = i; }
    }
    sv[tid] = bv; si[tid] = bi;
    __syncthreads();
    for (int off = T / 2; off > 0; off >>= 1) {
      if (tid < off) {
        float v2 = sv[tid + off]; int i2 = si[tid + off];
        float v1 = sv[tid];       int i1 = si[tid];
        bool take2 = (i1 < 0) || (i2 >= 0 && (v2 > v1 || (v2 == v1 && i2 < i1)));
        if (take2) { sv[tid] = v2; si[tid] = i2; }
      }
      __syncthreads();
    }
    if (tid == 0) { s_sum += sv[0]; s_pv = sv[0]; s_pi = si[0]; }
    __syncthreads();
  }
  if (tid == 0) {
    int k = (int)s_sum;                // trunc toward zero, as astype(int32)
    if (k < 1)  k = 1;
    if (k > 10) k = 10;
    s_k = k;
    float vld = gmeta[(size_t)bm * 4 + 3];
    mem_cnt[bm] = (vld > 0.5f) ? k : 0;
    s_pv = -3.0e38f; s_pi = -1;
  }
  __syncthreads();
  int K = s_k;

  // Phase 2: K min-extraction passes over cost (asc value, asc index).
  for (int t = 0; t < K; ++t) {
    float pv = s_pv; int pi = s_pi;
    float bv = 0.f;  int bi = -1;
    for (int i = tid; i < N_; i += T) {
      float v = cost_nm[((size_t)b * N_ + i) * M_ + m];
      bool elig = (v > pv) || (v == pv && i > pi);
      if (elig && (bi < 0 || v < bv || (v == bv && i < bi))) { bv = v; bi = i; }
    }
    sv[tid] = bv; si[tid] = bi;
    __syncthreads();
    for (int off = T / 2; off > 0; off >>= 1) {
      if (tid < off) {
        float v2 = sv[tid + off]; int i2 = si[tid + off];
        float v1 = sv[tid];       int i1 = si[tid];
        bool take2 = (i1 < 0) || (i2 >= 0 && (v2 < v1 || (v2 == v1 && i2 < i1)));
        if (take2) { sv[tid] = v2; si[tid] = i2; }
      }
      __syncthreads();
    }
    if (tid == 0) {
      mem_idx[bm * 16 + t]  = si[0];
      mem_cost[bm * 16 + t] = sv[0];
      s_pv = sv[0]; s_pi = si[0];
    }
    __syncthreads();
  }
}

// ---------------------------------------------------------------------------
// K5: per pred n: first-min over m of member costs -> (assigned, matched).
// ---------------------------------------------------------------------------
__global__ void k_assign(const int* __restrict__ mem_idx, const float* __restrict__ mem_cost,
                         const int* __restrict__ mem_cnt, float* __restrict__ out) {
  int bn = blockIdx.x * blockDim.x + threadIdx.x;
  if (bn >= B_ * N_) return;
  int b = bn / N_;
  int n = bn % N_;
  float best = BIGC; int best_m = -1;
  for (int m = 0; m < M_; ++m) {
    int bm = b * M_ + m;
    int cnt = mem_cnt[bm];
    for (int k = 0; k < cnt; ++k) {
      if (mem_idx[bm * 16 + k] == n) {
        float c = mem_cost[bm * 16 + k];
        if (c < best) { best = c; best_m = m; }   // strict < keeps first minimal m
        break;
      }
    }
  }
  bool assigned = best < BIGC;
  out[bn] = assigned ? 1.f : 0.f;
  out[(size_t)B_ * N_ + bn] = (float)(assigned ? best_m : -1);
}

// ---------------------------------------------------------------------------
extern "C" void kernel_launch(void* const* d_in, const int* in_sizes, int n_in,
                              void* d_out, int out_size, void* d_ws, size_t ws_size,
                              hipStream_t stream) {
  (void)in_sizes; (void)n_in; (void)out_size; (void)ws_size;
  const float* preds = (const float*)d_in[0];   // (B,N,78) f32
  const float* tgts  = (const float*)d_in[1];   // (B,M,78) f32
  const int*   masks = (const int*)d_in[2];     // (B,M) i32

  float* ws       = (float*)d_ws;
  float* gt_xs    = ws;                                   // B*M*S
  float* gmeta    = gt_xs + (size_t)B_ * M_ * S_;         // B*M*4
  float* pmeta    = gmeta + (size_t)B_ * M_ * 4;          // B*N*4
  float* cur_pred = pmeta + (size_t)B_ * N_ * 4;          // B*N*S
  float* part_nm  = cur_pred + (size_t)B_ * N_ * S_;      // B*N*M
  float* cost_nm  = part_nm + (size_t)B_ * N_ * M_;       // B*N*M
  float* iou_nm   = cost_nm + (size_t)B_ * N_ * M_;       // B*N*M
  float* mem_cost = iou_nm + (size_t)B_ * N_ * M_;        // B*M*16
  int*   mem_idx  = (int*)(mem_cost + (size_t)B_ * M_ * 16); // B*M*16
  int*   mem_cnt  = mem_idx + (size_t)B_ * M_ * 16;          // B*M

  k_gt<<<B_ * M_, 72, 0, stream>>>(tgts, masks, gt_xs, gmeta);
  k_pred<<<B_ * N_, 72, 0, stream>>>(preds, cur_pred, pmeta);
  k_dist_wmma<<<B_ * (N_ / 16), 32, 0, stream>>>(pmeta, gmeta, part_nm);
  int tot = B_ * N_ * M_;
  k_iou_cost<<<(tot + 255) / 256, 256, 0, stream>>>(cur_pred, gt_xs, gmeta, part_nm,
                                                    cost_nm, iou_nm);
  k_select<<<B_ * M_, 128, 0, stream>>>(iou_nm, cost_nm, gmeta, mem_idx, mem_cost, mem_cnt);
  k_assign<<<(B_ * N_ + 255) / 256, 256, 0, stream>>>(mem_idx, mem_cost, mem_cnt,
                                                      (float*)d_out);
}